// MambaVocGenerator_54116587930256
// MI455X (gfx1250) — compile-verified
//
#include <hip/hip_runtime.h>
#include <hip/hip_bf16.h>
#include <math.h>
#include <stdint.h>

// ---------------- problem constants (match reference) ----------------
#define DMODEL  256
#define DSTATE  64
#define DCONV   4
#define DINNER  512
#define DTRANK  16
#define NFFTQ   16
#define NBATCH  4
#define NMELS   80
#define NSEQ    2048
#define RROWS   (NBATCH * NSEQ)          // 8192 rows (b*SEQ + t)
#define KPRE    (NMELS * 7)              // 560, multiple of 4
#define NPOST   (NFFTQ + 2)              // 18 output channels
#define HALFCH  (NFFTQ / 2 + 1)          // 9

// GEMM tiling: block = 4 waves (128 thr) -> 64x16 output tile.
// Weight tile (16 x KC) staged in LDS via async global->LDS, double buffered.
#define KC      64                       // K chunk (divides 256 and 512)
#define LDSP    (KC + 4)                 // padded row pitch (bank-conflict free)
#define BUFF    (16 * LDSP)              // floats per LDS buffer
#define GEMM_LDS_BYTES (2 * BUFF * 4)    // 8704 B

typedef __attribute__((ext_vector_type(2))) float v2f;
typedef __attribute__((ext_vector_type(8))) float v8f;

__device__ __forceinline__ float sigmoid_f(float x) { return 1.f / (1.f + __expf(-x)); }

// async copy of one 16 x KC weight chunk into LDS buffer `buf`.
// 128 threads: thread t copies 8 consecutive K-floats of row n = t>>3
// as two B128 transfers.  2 async instructions per wave per chunk.
__device__ __forceinline__ void async_copy_wtile(
    const float* __restrict__ W, float* smem, int tn, int Kk, int chunk, int buf)
{
    const int n    = threadIdx.x >> 3;          // 0..15
    const int kpos = (threadIdx.x & 7) << 3;    // 0,8,...,56
    const float* g = W + (size_t)(tn + n) * Kk + chunk * KC + kpos;
    const uint32_t l = (uint32_t)(uintptr_t)(smem + buf * BUFF + n * LDSP + kpos);
    asm volatile("global_load_async_to_lds_b128 %0, %1, off"
                 :: "v"(l), "v"(g) : "memory");
    asm volatile("global_load_async_to_lds_b128 %0, %1, off"
                 :: "v"(l + 16u), "v"(g + 4) : "memory");
}

// =====================================================================
// fp32 WMMA GEMM:  C[R x N] = A[R x K] @ W[N x K]^T  (+bias)(+act)
// 4 waves/block stacked on M; W tile shared through LDS (async DMA,
// double buffered, counted s_wait_asynccnt).  K stepped by 4 via
// V_WMMA_F32_16X16X4_F32 (exact fp32 — GEMMs are a tiny share of the
// runtime, so no reason to drop to bf16).
//  A frag: lanes 0-15 -> M=lane, K={kk,kk+1}; lanes 16-31 -> K={kk+2,kk+3}
//  B frag: lanes 0-15 -> N=lane, same K split
//  C/D   : VGPR r -> row r + 8*(lane>=16), col lane%16
// Requires: R % 64 == 0, N % 16 == 0, K % KC == 0 (all true here).
// act: 0 = none, 1 = leaky-relu(0.01)
// =====================================================================
__global__ __launch_bounds__(128) void wmma_gemm_nt(
    const float* __restrict__ A, const float* __restrict__ W,
    const float* __restrict__ bias, float* __restrict__ C,
    int Rr, int Nn, int Kk, int act)
{
    extern __shared__ float smem[];
    const int lane = threadIdx.x & 31;
    const int wid  = threadIdx.x >> 5;
    const int ntn  = Nn >> 4;
    const int tm   = (blockIdx.x / ntn) * 64 + wid * 16;
    const int tn   = (blockIdx.x % ntn) << 4;
    const int l16  = lane & 15;
    const int koff = (lane >> 4) << 1;             // 0 (lanes 0-15) or 2 (lanes 16-31)

    const float* __restrict__ Ar = A + (size_t)(tm + l16) * Kk + koff;
    const int nch = Kk / KC;

    async_copy_wtile(W, smem, tn, Kk, 0, 0);       // prologue

    v8f acc = {0.f, 0.f, 0.f, 0.f, 0.f, 0.f, 0.f, 0.f};
    for (int c = 0; c < nch; ++c) {
        if (c + 1 < nch) {
            async_copy_wtile(W, smem, tn, Kk, c + 1, (c + 1) & 1);
            asm volatile("s_wait_asynccnt 0x2" ::: "memory");  // chunk c landed
        } else {
            asm volatile("s_wait_asynccnt 0x0" ::: "memory");
        }
        __syncthreads();                            // all waves' DMA visible

        const float* __restrict__ ac  = Ar + c * KC;
        const float* __restrict__ bsm = smem + (c & 1) * BUFF + l16 * LDSP + koff;
#pragma unroll 4
        for (int kk = 0; kk < KC; kk += 4) {
            v2f a; a[0] = ac[kk];  a[1] = ac[kk + 1];
            v2f b; b[0] = bsm[kk]; b[1] = bsm[kk + 1];
            acc = __builtin_amdgcn_wmma_f32_16x16x4_f32(false, a, false, b, (short)0,
                                                        acc, false, false);
        }
        __syncthreads();                            // done reading this buffer
    }

    const int rowbase = tm + ((lane >> 4) << 3);
    const float bv = bias ? bias[tn + l16] : 0.f;
    float* __restrict__ Cp = C + (size_t)rowbase * Nn + tn + l16;
#pragma unroll
    for (int r = 0; r < 8; ++r) {
        float v = acc[r] + bv;
        if (act == 1) v = (v >= 0.f) ? v : 0.01f * v;
        Cp[(size_t)r * Nn] = v;
    }
}

// =====================================================================
// conv_pre as WMMA GEMM with on-the-fly im2col (K = 80*7 = 560).
//  H[(b,t), o] = bias[o] + sum_{c,tap} x[b,c,t+tap-3] * Wpre[o,c,tap]
//  Output written directly in (b, t, c) layout -> fuses the transpose.
// =====================================================================
__global__ __launch_bounds__(128) void conv_pre_wmma(
    const float* __restrict__ x, const float* __restrict__ Wpre,
    const float* __restrict__ bias, float* __restrict__ H)
{
    const int lane = threadIdx.x & 31;
    const int wid  = threadIdx.x >> 5;
    const int tile = blockIdx.x * 4 + wid;
    const int ntn  = DMODEL >> 4;                  // 16
    const int tm   = (tile / ntn) << 4;
    const int tn   = (tile % ntn) << 4;
    const int l16  = lane & 15;
    const int koff = (lane >> 4) << 1;

    const int row = tm + l16;                      // same batch for whole tile
    const int bb  = row >> 11;
    const int t   = row & (NSEQ - 1);
    const float* __restrict__ xb = x + (size_t)bb * NMELS * NSEQ;
    const float* __restrict__ Wr = Wpre + (size_t)(tn + l16) * KPRE + koff;

    v8f acc = {0.f, 0.f, 0.f, 0.f, 0.f, 0.f, 0.f, 0.f};
    for (int kk = 0; kk < KPRE; kk += 4) {
        v2f a, b;
#pragma unroll
        for (int j = 0; j < 2; ++j) {
            const int k   = kk + koff + j;
            const int c   = k / 7;
            const int tap = k - c * 7;
            const int tt  = t + tap - 3;
            a[j] = (tt >= 0 && tt < NSEQ) ? xb[(size_t)c * NSEQ + tt] : 0.f;
        }
        b[0] = Wr[kk]; b[1] = Wr[kk + 1];
        acc = __builtin_amdgcn_wmma_f32_16x16x4_f32(false, a, false, b, (short)0, acc,
                                                    false, false);
    }

    const int rowbase = tm + ((lane >> 4) << 3);
    const float bv = bias[tn + l16];
    float* __restrict__ Hp = H + (size_t)rowbase * DMODEL + tn + l16;
#pragma unroll
    for (int r = 0; r < 8; ++r) Hp[(size_t)r * DMODEL] = acc[r] + bv;
}

// =====================================================================
// Causal depthwise conv (k=4, pad 3 left) + SiLU.  u = xz[:, :512]
// =====================================================================
__global__ __launch_bounds__(256) void dwconv_silu_kernel(
    const float* __restrict__ xz, const float* __restrict__ w,
    const float* __restrict__ b, float* __restrict__ uc)
{
    const int idx = blockIdx.x * blockDim.x + threadIdx.x;   // r*512 + d
    if (idx >= RROWS * DINNER) return;
    const int d  = idx & (DINNER - 1);
    const int r  = idx >> 9;
    const int t  = r & (NSEQ - 1);
    const int bb = r >> 11;
    float acc = b[d];
#pragma unroll
    for (int j = 0; j < DCONV; ++j) {
        const int tt = t - 3 + j;
        if (tt >= 0)
            acc += xz[((size_t)((bb << 11) + tt)) * (2 * DINNER) + d] * w[d * DCONV + j];
    }
    uc[idx] = acc * sigmoid_f(acc);                           // silu
}

// =====================================================================
// delta = softplus(x_dbl[:, :16] @ dt_proj_w^T + dt_proj_b)   (K=16 VALU)
// =====================================================================
__global__ __launch_bounds__(256) void delta_kernel(
    const float* __restrict__ xdbl, const float* __restrict__ dtw,
    const float* __restrict__ dtb, float* __restrict__ delta)
{
    const int idx = blockIdx.x * blockDim.x + threadIdx.x;   // r*512 + d
    if (idx >= RROWS * DINNER) return;
    const int d = idx & (DINNER - 1);
    const int r = idx >> 9;
    const float* __restrict__ xr = xdbl + (size_t)r * (DTRANK + 2 * DSTATE);
    const float* __restrict__ wr = dtw + (size_t)d * DTRANK;
    float acc = dtb[d];
#pragma unroll
    for (int k = 0; k < DTRANK; ++k) acc += xr[k] * wr[k];
    delta[idx] = (acc > 20.f) ? acc : log1pf(__expf(acc));   // softplus
}

// =====================================================================
// Selective scan, one wave32 per (batch, channel); each lane owns states
// n=lane and n=lane+32 in registers.  Recurrence chain per step is just
// v_mul -> v_exp -> v_fma; the C-reduction (5x shfl_xor) is off-path.
// Fuses +u*D and the silu(z) gate at the store.
// =====================================================================
__global__ __launch_bounds__(256) void scan_kernel(
    const float* __restrict__ delta, const float* __restrict__ uc,
    const float* __restrict__ xdbl,  const float* __restrict__ xz,
    const float* __restrict__ A_log, const float* __restrict__ D_skip,
    float* __restrict__ y)
{
    const int wave = (blockIdx.x * blockDim.x + threadIdx.x) >> 5;  // 0..2047
    const int lane = threadIdx.x & 31;
    if (wave >= NBATCH * DINNER) return;
    const int d  = wave & (DINNER - 1);
    const int bb = wave >> 9;

    const float A0 = -__expf(A_log[d * DSTATE + lane]);
    const float A1 = -__expf(A_log[d * DSTATE + lane + 32]);
    const float Dd = D_skip[d];

    const float* __restrict__ dl = delta + (size_t)bb * NSEQ * DINNER + d;
    const float* __restrict__ uu = uc    + (size_t)bb * NSEQ * DINNER + d;
    const float* __restrict__ zz = xz    + (size_t)bb * NSEQ * (2 * DINNER) + DINNER + d;
    const float* __restrict__ xd = xdbl  + (size_t)bb * NSEQ * (DTRANK + 2 * DSTATE);
    float* __restrict__ yp       = y     + (size_t)bb * NSEQ * DINNER + d;

    float h0 = 0.f, h1 = 0.f;
    for (int t = 0; t < NSEQ; ++t) {
        const float* __restrict__ xr = xd + (size_t)t * (DTRANK + 2 * DSTATE);
        const float dt_ = dl[(size_t)t * DINNER];
        const float ut  = uu[(size_t)t * DINNER];
        const float B0  = xr[DTRANK + lane];
        const float B1  = xr[DTRANK + lane + 32];
        const float C0  = xr[DTRANK + DSTATE + lane];
        const float C1  = xr[DTRANK + DSTATE + lane + 32];
        if (t + 1 < NSEQ)                                   // global_prefetch_b8
            __builtin_prefetch(xr + (DTRANK + 2 * DSTATE), 0, 1);

        const float du = dt_ * ut;
        h0 = __expf(dt_ * A0) * h0 + du * B0;               // state recurrence (chain)
        h1 = __expf(dt_ * A1) * h1 + du * B1;

        float yv = h0 * C0 + h1 * C1;                       // reduce over 64 states
#pragma unroll
        for (int off = 16; off > 0; off >>= 1) yv += __shfl_xor(yv, off, 32);

        if (lane == 0) {
            const float zv = zz[(size_t)t * (2 * DINNER)];
            yp[(size_t)t * DINNER] = (yv + ut * Dd) * (zv * sigmoid_f(zv));
        }
    }
}

// =====================================================================
// conv_post (256 -> 18, k=7, pad 3) on leaky-relu'd activations
// (leaky-relu fused into out_proj epilogue), then exp / sin.
// Output: spec (4,9,2048) then phase (4,9,2048), concatenated flat.
// =====================================================================
__global__ __launch_bounds__(256) void conv_post_kernel(
    const float* __restrict__ act, const float* __restrict__ W,
    const float* __restrict__ bias, float* __restrict__ dst)
{
    const int idx = blockIdx.x * blockDim.x + threadIdx.x;   // b*18*2048 + ch*2048 + t
    if (idx >= NBATCH * NPOST * NSEQ) return;
    const int t  = idx & (NSEQ - 1);
    const int ch = (idx >> 11) % NPOST;
    const int bb = idx / (NPOST * NSEQ);

    float acc = bias[ch];
#pragma unroll
    for (int tap = 0; tap < 7; ++tap) {
        const int tt = t + tap - 3;
        if (tt < 0 || tt >= NSEQ) continue;
        const float* __restrict__ row = act + (size_t)((bb << 11) + tt) * DMODEL;
        const float* __restrict__ wr  = W + (size_t)ch * DMODEL * 7 + tap;
        for (int c = 0; c < DMODEL; ++c) acc += row[c] * wr[c * 7];
    }
    if (ch < HALFCH)
        dst[(size_t)bb * HALFCH * NSEQ + ch * NSEQ + t] = __expf(acc);
    else
        dst[(size_t)NBATCH * HALFCH * NSEQ +
            (size_t)bb * HALFCH * NSEQ + (ch - HALFCH) * NSEQ + t] = __sinf(acc);
}

// =====================================================================
extern "C" void kernel_launch(void* const* d_in, const int* in_sizes, int n_in,
                              void* d_out, int out_size, void* d_ws, size_t ws_size,
                              hipStream_t stream)
{
    const float* x          = (const float*)d_in[0];
    const float* conv_pre_w = (const float*)d_in[1];
    const float* conv_pre_b = (const float*)d_in[2];
    const float* in_proj_w  = (const float*)d_in[3];
    const float* dw_conv_w  = (const float*)d_in[4];
    const float* dw_conv_b  = (const float*)d_in[5];
    const float* x_proj_w   = (const float*)d_in[6];
    const float* dt_proj_w  = (const float*)d_in[7];
    const float* dt_proj_b  = (const float*)d_in[8];
    const float* A_log      = (const float*)d_in[9];
    const float* D_skip     = (const float*)d_in[10];
    const float* out_proj_w = (const float*)d_in[11];
    const float* conv_post_w= (const float*)d_in[12];
    const float* conv_post_b= (const float*)d_in[13];
    float* out = (float*)d_out;

    // workspace partition (floats)
    float* ws   = (float*)d_ws;
    float* ht   = ws;                                 // 8192*256
    float* xz   = ht   + (size_t)RROWS * DMODEL;      // 8192*1024
    float* uc   = xz   + (size_t)RROWS * 2 * DINNER;  // 8192*512
    float* xdbl = uc   + (size_t)RROWS * DINNER;      // 8192*144
    float* dlt  = xdbl + (size_t)RROWS * (DTRANK + 2 * DSTATE); // 8192*512
    float* yg   = dlt  + (size_t)RROWS * DINNER;      // 8192*512
    float* oact = yg   + (size_t)RROWS * DINNER;      // 8192*256

    // 1) conv_pre (WMMA + im2col, fused bias + transpose) -> ht [R x 256]
    {
        const int tiles = (RROWS / 16) * (DMODEL / 16);       // 8192
        conv_pre_wmma<<<tiles / 4, 128, 0, stream>>>(x, conv_pre_w, conv_pre_b, ht);
    }
    // 2) in_proj: xz = ht @ in_proj_w^T  [R x 1024]   (async-LDS WMMA GEMM)
    {
        const int blocks = (RROWS / 64) * ((2 * DINNER) / 16); // 8192
        wmma_gemm_nt<<<blocks, 128, GEMM_LDS_BYTES, stream>>>(ht, in_proj_w, nullptr, xz,
                                                              RROWS, 2 * DINNER, DMODEL, 0);
    }
    // 3) depthwise causal conv + silu -> uc [R x 512]
    {
        const int n = RROWS * DINNER;
        dwconv_silu_kernel<<<(n + 255) / 256, 256, 0, stream>>>(xz, dw_conv_w, dw_conv_b, uc);
    }
    // 4) x_proj: xdbl = uc @ x_proj_w^T  [R x 144]
    {
        const int blocks = (RROWS / 64) * ((DTRANK + 2 * DSTATE) / 16); // 1152
        wmma_gemm_nt<<<blocks, 128, GEMM_LDS_BYTES, stream>>>(uc, x_proj_w, nullptr, xdbl,
                                                              RROWS, DTRANK + 2 * DSTATE,
                                                              DINNER, 0);
    }
    // 5) delta = softplus(dt @ dt_proj_w^T + b) [R x 512]
    {
        const int n = RROWS * DINNER;
        delta_kernel<<<(n + 255) / 256, 256, 0, stream>>>(xdbl, dt_proj_w, dt_proj_b, dlt);
    }
    // 6) selective scan (+ D skip + silu(z) gate) -> yg [R x 512]
    {
        const int threads = NBATCH * DINNER * 32;             // 2048 waves
        scan_kernel<<<threads / 256, 256, 0, stream>>>(dlt, uc, xdbl, xz, A_log, D_skip, yg);
    }
    // 7) out_proj with fused leaky-relu -> oact [R x 256]
    {
        const int blocks = (RROWS / 64) * (DMODEL / 16);      // 2048
        wmma_gemm_nt<<<blocks, 128, GEMM_LDS_BYTES, stream>>>(yg, out_proj_w, nullptr, oact,
                                                              RROWS, DMODEL, DINNER, 1);
    }
    // 8) conv_post + exp/sin -> d_out
    {
        const int n = NBATCH * NPOST * NSEQ;                  // 147456
        conv_post_kernel<<<(n + 255) / 256, 256, 0, stream>>>(oact, conv_post_w,
                                                              conv_post_b, out);
    }
}